// AttentionAggregation_62053687492655
// MI455X (gfx1250) — compile-verified
//
#include <hip/hip_runtime.h>

// ---------------------------------------------------------------------------
// AttentionAggregation for MI455X (gfx1250, wave32)
//   v = x @ W_v.T          -> V_WMMA_F32_16X16X4_F32, k-pair-interleaved W^T
//                             in LDS (ds_load_b64 B fragments, conflict-free),
//                             2 row-stripes per wave to amortize B loads
//   y = segsum(alpha*v[j]) -> coalesced L2 gather + run-length atomics
// ---------------------------------------------------------------------------

typedef __attribute__((ext_vector_type(2))) float v2f;
typedef __attribute__((ext_vector_type(8))) float v8f;

#define F 128
#define KP 64              // number of k-pairs (F/2)
#define BSTRIDE 144        // float2 per k-pair row: 128 + 16 pad
                           // -> dword-bank shift between k-pair rows = (2*144)%64 = 32:
                           //    lanes 0-15 cover banks {2c..2c+31}, lanes 16-31 {+32}
                           //    => one ds_load_b64 touches all 64 banks exactly once
#define GEMM_BLOCK 256
#define WAVES_PER_BLOCK (GEMM_BLOCK / 32)
#define PAIRS_PER_BLOCK 256

// ------------------------- Kernel 1: v = x @ W^T ---------------------------
// One wave32 computes a 32-row x 128-col super-stripe of v:
//   2 stripes x 8 accumulator tiles (16x16 f32), K loop of 32 wmma steps,
//   each B fragment (one ds_load_b64) feeds 2 wmma ops.
__global__ __launch_bounds__(GEMM_BLOCK)
void attn_gemm_wmma(const float* __restrict__ x, const float* __restrict__ W,
                    float* __restrict__ v, int n_rows) {
    // WP[kp][o] = { W^T[2kp][o], W^T[2kp+1][o] } = { W[o][2kp], W[o][2kp+1] }
    __shared__ v2f WP[KP * BSTRIDE];   // 73.7 KB of 320 KB

    for (int e = threadIdx.x; e < KP * F; e += GEMM_BLOCK) {
        const int o  = e >> 6;          // 0..127 (row of W)
        const int kp = e & (KP - 1);    // 0..63  (k-pair) -> coalesced 8B reads of W
        WP[kp * BSTRIDE + o] = *(const v2f*)(W + (size_t)o * F + 2 * kp);
    }
    __syncthreads();

    const int lane = threadIdx.x & 31;
    const int wave = threadIdx.x >> 5;
    const int lo   = lane & 15;   // M (A) / N (B,C) index
    const int hi   = lane >> 4;   // selects K pair (A,B) / row half (C)
    const int nsuper = n_rows >> 5;   // 32-row super-stripes

    for (int s = blockIdx.x * WAVES_PER_BLOCK + wave; s < nsuper;
         s += gridDim.x * WAVES_PER_BLOCK) {
        const int row0 = s << 5;

        v8f acc[2][8];
#pragma unroll
        for (int m = 0; m < 2; ++m)
#pragma unroll
            for (int n = 0; n < 8; ++n)
                acc[m][n] = (v8f){0.f, 0.f, 0.f, 0.f, 0.f, 0.f, 0.f, 0.f};

        // A fragment (ISA 16x4 f32): lanes 0-15 -> M=lane, K={k4,k4+1};
        // lanes 16-31 -> M=lane-16, K={k4+2,k4+3}. Row-major x -> float2 load.
        const float* xrow0 = x + (size_t)(row0 + lo) * F + 2 * hi;
        const float* xrow1 = xrow0 + (size_t)16 * F;

#pragma unroll 2
        for (int k4 = 0; k4 < F; k4 += 4) {
            const v2f a0 = *(const v2f*)(xrow0 + k4);
            const v2f a1 = *(const v2f*)(xrow1 + k4);
            const v2f* bp = &WP[((k4 >> 1) + hi) * BSTRIDE + lo];
#pragma unroll
            for (int n = 0; n < 8; ++n) {
                // B fragment (4x16): one aligned b64 = rows {k4+2hi, k4+2hi+1}
                const v2f b = bp[n * 16];
                acc[0][n] = __builtin_amdgcn_wmma_f32_16x16x4_f32(
                    false, a0, false, b, (short)0, acc[0][n], false, false);
                acc[1][n] = __builtin_amdgcn_wmma_f32_16x16x4_f32(
                    false, a1, false, b, (short)0, acc[1][n], false, false);
            }
        }

        // C layout: VGPR r -> row (r + 8*hi), col (n*16 + lo)
#pragma unroll
        for (int m = 0; m < 2; ++m) {
#pragma unroll
            for (int n = 0; n < 8; ++n) {
                float* out = v + (size_t)(row0 + 16 * m + 8 * hi) * F + n * 16 + lo;
#pragma unroll
                for (int r = 0; r < 8; ++r)
                    out[(size_t)r * F] = acc[m][n][r];
            }
        }
    }
}

// ------------------------- Kernel 2: y = 0 ---------------------------------
__global__ __launch_bounds__(256)
void attn_zero(float4* __restrict__ y, long n4) {
    const long i = (long)blockIdx.x * 256 + threadIdx.x;
    if (i < n4) y[i] = make_float4(0.f, 0.f, 0.f, 0.f);
}

// ------------------ Kernel 3: y[i] += alpha * v[j] -------------------------
// 128 threads = one feature each. Block owns 256 consecutive (sorted-by-i)
// pairs; accumulate runs of equal idx_i in registers, flush with f32 atomics.
__global__ __launch_bounds__(128)
void attn_scatter(const float* __restrict__ v, const float* __restrict__ alpha,
                  const int* __restrict__ idx_i, const int* __restrict__ idx_j,
                  float* __restrict__ y, int n_pairs) {
    __shared__ int   s_i[PAIRS_PER_BLOCK];
    __shared__ int   s_j[PAIRS_PER_BLOCK];
    __shared__ float s_a[PAIRS_PER_BLOCK];

    const int p0  = blockIdx.x * PAIRS_PER_BLOCK;
    const int cnt = min(PAIRS_PER_BLOCK, n_pairs - p0);
    if (cnt <= 0) return;

    for (int q = threadIdx.x; q < cnt; q += 128) {
        s_i[q] = idx_i[p0 + q];
        s_j[q] = idx_j[p0 + q];
        s_a[q] = alpha[p0 + q];
    }
    __syncthreads();

    const int t = threadIdx.x;     // feature index 0..127
    float acc = 0.f;
    int   cur = s_i[0];

    for (int q = 0; q < cnt; ++q) {
        const int i = s_i[q];      // uniform across block -> scalar branch
        if (i != cur) {
            atomicAdd(&y[(size_t)cur * F + t], acc);
            acc = 0.f;
            cur = i;
        }
        acc = fmaf(s_a[q], v[(size_t)s_j[q] * F + t], acc);  // coalesced 512B gather (L2-resident)
    }
    atomicAdd(&y[(size_t)cur * F + t], acc);
}

// ---------------------------------------------------------------------------
extern "C" void kernel_launch(void* const* d_in, const int* in_sizes, int n_in,
                              void* d_out, int out_size, void* d_ws, size_t ws_size,
                              hipStream_t stream) {
    (void)n_in; (void)out_size; (void)ws_size;

    const float* x     = (const float*)d_in[0];
    const float* alpha = (const float*)d_in[1];
    const int*   idx_i = (const int*)d_in[2];
    const int*   idx_j = (const int*)d_in[3];
    const float* Wv    = (const float*)d_in[4];
    float* y = (float*)d_out;
    float* v = (float*)d_ws;                 // 100000*128*4 = 51.2 MB scratch

    const int n_nodes = in_sizes[0] / F;     // 100000 (divisible by 32)
    const int n_pairs = in_sizes[1];         // 640000

    // y = 0 (harness poisons d_out)
    const long n4 = (long)n_nodes * F / 4;
    attn_zero<<<(int)((n4 + 255) / 256), 256, 0, stream>>>((float4*)y, n4);

    // v = x @ W^T
    const int nsuper  = n_nodes / 32;
    const int gblocks = (nsuper + WAVES_PER_BLOCK - 1) / WAVES_PER_BLOCK;
    attn_gemm_wmma<<<gblocks, GEMM_BLOCK, 0, stream>>>(x, Wv, v, n_nodes);

    // y[idx_i] += alpha * v[idx_j]
    const int sblocks = (n_pairs + PAIRS_PER_BLOCK - 1) / PAIRS_PER_BLOCK;
    attn_scatter<<<sblocks, 128, 0, stream>>>(v, alpha, idx_i, idx_j, y, n_pairs);
}